// StreamingAttentionSink_42417097015344
// MI455X (gfx1250) — compile-verified
//
#include <hip/hip_runtime.h>
#include <hip/hip_bf16.h>
#include <math.h>

#define NB      8          // batch
#define HQ      32
#define HKV     8
#define GG      4          // GQA group = HQ/HKV
#define DD      128        // head dim
#define BLKSZ   16
#define BPS     289        // blocks per sequence
#define NTILE   256        // 16-token tiles covering 4086 tokens (last tile: 5 past + 1 cur)
#define NSPLIT  8
#define TPS     (NTILE / NSPLIT)
#define KSTR    132        // padded LDS row stride (floats) for K/V tiles
#define SCALE   0.08838834764831845f  // 1/sqrt(128)
#define NEGINF  (-3.0e38f)

// workspace layout in floats
#define QROT_OFF 0
#define KROT_OFF (QROT_OFF + NB*HQ*DD)               // 32768
#define TAB_OFF  (KROT_OFF + NB*HKV*DD)              // 40960
#define TAB_N    (4086*64)
#define PART_STRIDE (GG*DD + 2*GG)                   // 520
#define PART_OFF (TAB_OFF + TAB_N*2)                 // 563968
#define ATTN_OFF (PART_OFF + NB*HKV*NSPLIT*PART_STRIDE)
#define KSPLIT   4
#define OUT_N    (NB*HQ*DD)                          // 32768 output elements
#define GEMM_OFF (ATTN_OFF + OUT_N)

typedef __attribute__((ext_vector_type(2))) float v2f;
typedef __attribute__((ext_vector_type(8))) float v8f;
typedef __attribute__((ext_vector_type(4))) int   v4i;

#if defined(__has_builtin)
#if __has_builtin(__builtin_amdgcn_global_load_async_to_lds_b128)
#define HAS_ASYNC_LDS 1
#endif
#endif

#ifdef HAS_ASYNC_LDS
typedef __attribute__((address_space(1))) v4i gv4i;   // global (device) int4
typedef __attribute__((address_space(3))) v4i lv4i;   // LDS int4
#endif

__device__ __forceinline__ void g2l_b128(float* lds, const float* g) {
#ifdef HAS_ASYNC_LDS
  __builtin_amdgcn_global_load_async_to_lds_b128((gv4i*)g, (lv4i*)lds, 0, 0);
#else
  *(float4*)lds = *(const float4*)g;
#endif
}

__device__ __forceinline__ void wait_async_lds() {
#ifdef HAS_ASYNC_LDS
#if __has_builtin(__builtin_amdgcn_s_wait_asynccnt)
  __builtin_amdgcn_s_wait_asynccnt(0);
#else
  asm volatile("s_wait_asynccnt 0" ::: "memory");
#endif
#endif
}

// ---------------- Kernel 1: RoPE for current-token q and k (pos = 4095) -------------
__global__ void k_rope_qk(const float* __restrict__ q, const float* __restrict__ k,
                          float* __restrict__ ws) {
  int hid = blockIdx.x;          // 0..255 = q heads, 256..319 = k heads
  int i = threadIdx.x;           // 0..63 (rotation pair index)
  const float* src; float* dst;
  if (hid < NB*HQ) { src = q + (size_t)hid*DD; dst = ws + QROT_OFF + (size_t)hid*DD; }
  else { int id = hid - NB*HQ; src = k + (size_t)id*DD; dst = ws + KROT_OFF + (size_t)id*DD; }
  float inv = __expf(-(float)i * (9.210340371976184f / 64.0f)); // 10000^(-i/64)
  float fr = 4095.0f * inv;
  float s, c; __sincosf(fr, &s, &c);
  float x1 = src[i], x2 = src[i + 64];
  dst[i]      = x1*c - x2*s;
  dst[i + 64] = x2*c + x1*s;
}

// ---------------- Kernel 2: cos/sin table for streamed past positions --------------
__global__ void k_rope_tab(float* __restrict__ ws) {
  int tid = blockIdx.x * blockDim.x + threadIdx.x;
  if (tid >= TAB_N) return;
  int t = tid >> 6, j = tid & 63;
  int pos = (t < 16) ? t : (t + 10);    // sink: 0..15 ; window: 26..4094
  float inv = __expf(-(float)j * (9.210340371976184f / 64.0f));
  float s, c; __sincosf((float)pos * inv, &s, &c);
  ws[TAB_OFF + 2*tid]     = c;
  ws[TAB_OFF + 2*tid + 1] = s;
}

// ---------------- Kernel 3: flash-decode partials, 1 wave per (b,kvh,split) --------
__global__ __launch_bounds__(32)
void k_attn_partial(const float* __restrict__ vcur, const int* __restrict__ bt,
                    const float* __restrict__ kcache, const float* __restrict__ vcache,
                    float* __restrict__ ws) {
  const int wg    = blockIdx.x;
  const int split = wg % NSPLIT;
  const int kvh   = (wg / NSPLIT) % HKV;
  const int b     = wg / (NSPLIT * HKV);
  const int lane  = threadIdx.x;

  __shared__ __align__(16) float Qs[GG*DD];
  __shared__ __align__(16) float Ks[2][BLKSZ][KSTR];
  __shared__ __align__(16) float Vs[2][BLKSZ][KSTR];
  __shared__ float Ps[GG][BLKSZ];

  const float* qrot = ws + QROT_OFF + ((size_t)(b*HQ + kvh*GG))*DD;
  for (int idx = lane*4; idx < GG*DD; idx += 128)
    *(float4*)&Qs[idx] = *(const float4*)&qrot[idx];

  float m[GG], l[GG], acc[GG][4];
#pragma unroll
  for (int g = 0; g < GG; ++g) {
    m[g] = NEGINF; l[g] = 0.f;
    acc[g][0]=acc[g][1]=acc[g][2]=acc[g][3]=0.f;
  }

  const int t0tile = split * TPS;

  auto load_tile = [&](int buf, int ti) {
    int pb = bt[b*BPS + ((ti == 0) ? 0 : (33 + ti))];
    int nload = (ti == NTILE-1) ? 5 : BLKSZ;
    const float* kb = kcache + ((size_t)pb*BLKSZ*HKV + kvh) * DD;
    const float* vb = vcache + ((size_t)pb*BLKSZ*HKV + kvh) * DD;
    for (int r = 0; r < nload; ++r) {
      g2l_b128(&Ks[buf][r][lane*4], kb + (size_t)r*HKV*DD + lane*4);
      g2l_b128(&Vs[buf][r][lane*4], vb + (size_t)r*HKV*DD + lane*4);
    }
  };

  load_tile(0, t0tile);

  for (int i = 0; i < TPS; ++i) {
    const int ti  = t0tile + i;
    const int buf = i & 1;
    wait_async_lds();
    __syncthreads();
    if (i + 1 < TPS) load_tile((i + 1) & 1, ti + 1);   // prefetch next tile

    const bool last  = (ti == NTILE - 1);
    const int  nrope = last ? 5 : BLKSZ;
    int n = nrope;

    // re-RoPE the cached keys in LDS (stream positions from table)
    for (int idx = lane; idx < nrope*64; idx += 32) {
      int r = idx >> 6, j = idx & 63;
      int t = ti*BLKSZ + r;
      float c = ws[TAB_OFF + 2*(t*64 + j)];
      float s = ws[TAB_OFF + 2*(t*64 + j) + 1];
      float x1 = Ks[buf][r][j], x2 = Ks[buf][r][j + 64];
      Ks[buf][r][j]      = x1*c - x2*s;
      Ks[buf][r][j + 64] = x2*c + x1*s;
    }
    if (last) {  // append current token (k already RoPE'd at pos 4095; v raw)
      const float* kc = ws + KROT_OFF + ((size_t)(b*HKV + kvh))*DD;
      const float* vc = vcur + ((size_t)(b*HKV + kvh))*DD;
      *(float4*)&Ks[buf][5][lane*4] = *(const float4*)&kc[lane*4];
      *(float4*)&Vs[buf][5][lane*4] = *(const float4*)&vc[lane*4];
      n = 6;
    }
    __syncthreads();

    // scores: lane pair (token = lane&15, dim-half = lane>>4)
    const int t  = lane & 15;
    const int jb = (lane >> 4) * 64;
    float sg[GG] = {0.f, 0.f, 0.f, 0.f};
    for (int j = 0; j < 64; j += 4) {
      float4 kv = *(const float4*)&Ks[buf][t][jb + j];
#pragma unroll
      for (int g = 0; g < GG; ++g) {
        float4 qq = *(const float4*)&Qs[g*DD + jb + j];
        sg[g] += qq.x*kv.x + qq.y*kv.y + qq.z*kv.z + qq.w*kv.w;
      }
    }
#pragma unroll
    for (int g = 0; g < GG; ++g) {
      sg[g] += __shfl_xor(sg[g], 16, 32);              // combine dim halves
      sg[g] = (t < n) ? sg[g]*SCALE : NEGINF;
    }

    // online softmax update (wave-uniform m/l after butterflies)
#pragma unroll
    for (int g = 0; g < GG; ++g) {
      float mx = sg[g];
      mx = fmaxf(mx, __shfl_xor(mx, 1, 32));
      mx = fmaxf(mx, __shfl_xor(mx, 2, 32));
      mx = fmaxf(mx, __shfl_xor(mx, 4, 32));
      mx = fmaxf(mx, __shfl_xor(mx, 8, 32));
      float mn    = fmaxf(m[g], mx);
      float alpha = __expf(m[g] - mn);
      float p     = __expf(sg[g] - mn);
      float rs = p;
      rs += __shfl_xor(rs, 1, 32);
      rs += __shfl_xor(rs, 2, 32);
      rs += __shfl_xor(rs, 4, 32);
      rs += __shfl_xor(rs, 8, 32);
      l[g] = l[g]*alpha + rs;
      m[g] = mn;
      acc[g][0]*=alpha; acc[g][1]*=alpha; acc[g][2]*=alpha; acc[g][3]*=alpha;
      if (lane < 16) Ps[g][t] = p;
    }
    __syncthreads();

    // P @ V : lane owns dims [lane*4, lane*4+4)
    for (int tt = 0; tt < n; ++tt) {
      float4 vv = *(const float4*)&Vs[buf][tt][lane*4];
#pragma unroll
      for (int g = 0; g < GG; ++g) {
        float pg = Ps[g][tt];
        acc[g][0] += pg*vv.x; acc[g][1] += pg*vv.y;
        acc[g][2] += pg*vv.z; acc[g][3] += pg*vv.w;
      }
    }
    __syncthreads();
  }

  float* pbase = ws + PART_OFF + (size_t)wg * PART_STRIDE;
#pragma unroll
  for (int g = 0; g < GG; ++g) {
    pbase[g*DD + lane*4 + 0] = acc[g][0];
    pbase[g*DD + lane*4 + 1] = acc[g][1];
    pbase[g*DD + lane*4 + 2] = acc[g][2];
    pbase[g*DD + lane*4 + 3] = acc[g][3];
  }
  if (lane == 0) {
#pragma unroll
    for (int g = 0; g < GG; ++g) {
      pbase[GG*DD + g]      = m[g];
      pbase[GG*DD + GG + g] = l[g];
    }
  }
}

// ---------------- Kernel 4: merge splits (log-sum-exp) -> attn (B,32,128) ----------
__global__ void k_attn_reduce(float* __restrict__ ws) {
  int bh = blockIdx.x;                 // b*32 + h
  int b = bh >> 5, h = bh & 31;
  int kvh = h >> 2, g = h & 3;
  int d = threadIdx.x;                 // 0..127
  const float* pw = ws + PART_OFF;
  float ms[NSPLIT];
  float M = NEGINF;
#pragma unroll
  for (int s = 0; s < NSPLIT; ++s) {
    ms[s] = pw[((size_t)((b*HKV + kvh)*NSPLIT + s))*PART_STRIDE + GG*DD + g];
    M = fmaxf(M, ms[s]);
  }
  float L = 0.f, o = 0.f;
#pragma unroll
  for (int s = 0; s < NSPLIT; ++s) {
    const float* pb2 = pw + ((size_t)((b*HKV + kvh)*NSPLIT + s))*PART_STRIDE;
    float w = __expf(ms[s] - M);
    L += w * pb2[GG*DD + GG + g];
    o += w * pb2[g*DD + d];
  }
  ws[ATTN_OFF + (size_t)bh*DD + d] = o / L;
}

// ------- Kernel 5: partial GEMM  part[ks] = attn(8x1024-slice) @ W_o slice ---------
// grid = KSPLIT * 64 blocks, 128 threads (4 waves); WMMA f32 16x16x4,
// double-buffered async-LDS staging; A padded to 16 rows (8..15 zero).
#define WSTR 72            // padded LDS stride (floats): conflict-free fragment reads
#define KCH  64            // K-chunk
#define CHUNKS_PER_KS (4096 / KSPLIT / KCH)   // 16
__global__ __launch_bounds__(128)
void k_gemm_wo(const float* __restrict__ Wo, float* __restrict__ ws) {
  const int ks   = blockIdx.x >> 6;            // 0..3  K split
  const int cblk = (blockIdx.x & 63) * 64;     // 64-column tile
  const int tid  = threadIdx.x;
  const int lane = tid & 31;
  const int wv   = tid >> 5;                   // wave 0..3, each owns 16 columns
  const int c0   = wv * 16;
  const int kbase = ks * (4096 / KSPLIT);

  __shared__ __align__(16) float Wt[2][KCH][WSTR];
  __shared__ __align__(16) float At[2][16][WSTR];

  const float* attn = ws + ATTN_OFF;

  { // zero the A padding rows (8..15) of both buffers, once
    int idx = tid * 4;                         // 0..508
    int r = 8 + (idx >> 6), c = idx & 63;
    float4 z = {0.f, 0.f, 0.f, 0.f};
    *(float4*)&At[0][r][c] = z;
    *(float4*)&At[1][r][c] = z;
  }

  auto stage = [&](int buf, int kk) {
    { // A chunk 8x64 (async, L2-resident)
      int idx = tid * 4;
      int r = idx >> 6, c = idx & 63;
      g2l_b128(&At[buf][r][c], attn + (size_t)r*4096 + kk + c);
    }
    { // W chunk 64x64 (async)
      int r0 = tid >> 4;
      int c  = (tid & 15) * 4;
#pragma unroll
      for (int p = 0; p < 8; ++p) {
        int r = r0 + p*8;
        g2l_b128(&Wt[buf][r][c], Wo + (size_t)(kk + r)*4096 + cblk + c);
      }
    }
  };

  v8f acc = {0.f,0.f,0.f,0.f,0.f,0.f,0.f,0.f};
  stage(0, kbase);

  const int row  = lane & 15;                  // A fragment M-row (8..15 hit zero pad)
  const int koff = (lane < 16) ? 0 : 2;        // A/B VGPR0={K0|K2}, VGPR1={K1|K3}

  for (int it = 0; it < CHUNKS_PER_KS; ++it) {
    const int buf = it & 1;
    wait_async_lds();
    __syncthreads();
    if (it + 1 < CHUNKS_PER_KS) stage(buf ^ 1, kbase + (it + 1)*KCH);

#pragma unroll
    for (int k0 = 0; k0 < KCH; k0 += 4) {
      v2f a = *(const v2f*)&At[buf][row][k0 + koff];   // unconditional ds_load_b64
      v2f bf;
      bf.x = Wt[buf][k0 + koff][c0 + row];
      bf.y = Wt[buf][k0 + koff + 1][c0 + row];
      acc = __builtin_amdgcn_wmma_f32_16x16x4_f32(false, a, false, bf,
                                                  (short)0, acc, false, false);
    }
    __syncthreads();
  }

  float* part = ws + GEMM_OFF + (size_t)ks * OUT_N;
  if (lane < 16) {
    int col = cblk + c0 + lane;
#pragma unroll
    for (int r = 0; r < 8; ++r)        // C layout: VGPR r holds M=r for lanes 0..15
      part[(size_t)r*4096 + col] = acc[r];
  }
}

// ---------------- Kernel 6: sum the K-split partials (deterministic order) ---------
__global__ void k_gemm_reduce(const float* __restrict__ ws, float* __restrict__ out) {
  int i = blockIdx.x * blockDim.x + threadIdx.x;
  if (i >= OUT_N) return;
  const float* p = ws + GEMM_OFF;
  float s = 0.f;
#pragma unroll
  for (int ksp = 0; ksp < KSPLIT; ++ksp) s += p[(size_t)ksp * OUT_N + i];
  out[i] = s;
}

// -----------------------------------------------------------------------------------
extern "C" void kernel_launch(void* const* d_in, const int* in_sizes, int n_in,
                              void* d_out, int out_size, void* d_ws, size_t ws_size,
                              hipStream_t stream) {
  (void)in_sizes; (void)n_in; (void)out_size; (void)ws_size;
  const float* q  = (const float*)d_in[0];
  const float* k  = (const float*)d_in[1];
  const float* v  = (const float*)d_in[2];
  const int*   bt = (const int*)d_in[5];
  const float* kc = (const float*)d_in[6];
  const float* vc = (const float*)d_in[7];
  const float* Wo = (const float*)d_in[8];
  float* ws  = (float*)d_ws;
  float* out = (float*)d_out;

  k_rope_qk     <<<NB*(HQ+HKV), 64, 0, stream>>>(q, k, ws);
  k_rope_tab    <<<(TAB_N + 255)/256, 256, 0, stream>>>(ws);
  k_attn_partial<<<NB*HKV*NSPLIT, 32, 0, stream>>>(v, bt, kc, vc, ws);
  k_attn_reduce <<<NB*HQ, 128, 0, stream>>>(ws);
  k_gemm_wo     <<<KSPLIT*64, 128, 0, stream>>>(Wo, ws);
  k_gemm_reduce <<<(OUT_N + 255)/256, 256, 0, stream>>>(ws, out);
}